// NonIsotropicvMFHead_30906584662531
// MI455X (gfx1250) — compile-verified
//
#include <hip/hip_runtime.h>
#include <hip/hip_bf16.h>
#include <math.h>

typedef float v2f __attribute__((ext_vector_type(2)));
typedef float v8f __attribute__((ext_vector_type(8)));

#define Bsz 64
#define Fdim 128
#define Cdim 1000
#define Cpad 1008
#define Hdim 256
#define NMC 8
#define Mrows (NMC*Bsz)   // 512

// ---------- helpers ----------
__device__ __forceinline__ float wave_sum(float x) {
  #pragma unroll
  for (int m = 16; m > 0; m >>= 1) x += __shfl_xor(x, m, 32);
  return x;
}

__device__ __forceinline__ float block_sum_128(float v, float* sd) {
  int t = threadIdx.x;
  sd[t] = v; __syncthreads();
  for (int s = 64; s > 0; s >>= 1) { if (t < s) sd[t] += sd[t + s]; __syncthreads(); }
  float r = sd[0]; __syncthreads();
  return r;
}

__device__ __forceinline__ unsigned hashu(unsigned x) {
  x ^= x >> 17; x *= 0xed5ad4bbu;
  x ^= x >> 11; x *= 0xac4c1b51u;
  x ^= x >> 15; x *= 0x31848babu;
  x ^= x >> 14; return x;
}
__device__ __forceinline__ float u01(unsigned x) {
  // (0,1]
  return ((hashu(x) >> 8) + 1u) * (1.0f / 16777216.0f);
}

// ---------- 1) row-normalize features ----------
__global__ void k_norm_rows(const float* __restrict__ X, float* __restrict__ Y) {
  __shared__ float sd[128];
  int b = blockIdx.x, d = threadIdx.x;
  float x = X[b * Fdim + d];
  float s = block_sum_128(x * x, sd);
  Y[b * Fdim + d] = x / fmaxf(sqrtf(s), 1e-12f);
}

// ---------- 2) MLP layer: Out = relu(X @ W^T + bias), via f32 WMMA ----------
__global__ void k_mlp_wmma(const float* __restrict__ X, const float* __restrict__ W,
                           const float* __restrict__ bias, float* __restrict__ Out,
                           int M, int N, int K, int tilesN) {
  int wave = (blockIdx.x * blockDim.x + threadIdx.x) >> 5;
  int lane = threadIdx.x & 31;
  int tm = wave / tilesN, tn = wave % tilesN;
  int m0 = tm * 16, n0 = tn * 16;
  if (m0 >= M) return;
  int row = lane & 15, hi = lane >> 4;
  const float* ap = X + (m0 + row) * K + hi * 2;   // A: 16xK rows of X
  const float* bp = W + (n0 + row) * K + hi * 2;   // B: columns = rows of W
  v8f acc = {};
  for (int k0 = 0; k0 < K; k0 += 4) {
    v2f a, b;
    a.x = ap[k0]; a.y = ap[k0 + 1];
    b.x = bp[k0]; b.y = bp[k0 + 1];
    acc = __builtin_amdgcn_wmma_f32_16x16x4_f32(false, a, false, b, (short)0, acc,
                                                false, false);
  }
  int nn = n0 + (lane & 15);
  int mb = m0 + hi * 8;
  float bv = bias[nn];
  #pragma unroll
  for (int v = 0; v < 8; v++) {
    float val = fmaxf(acc[v] + bv, 0.0f);
    Out[(mb + v) * N + nn] = val;
  }
}

// ---------- 3) final MLP layer -> kappa (softplus + eps) ----------
__global__ void k_kappa(const float* __restrict__ h1, const float* __restrict__ W2,
                        const float* __restrict__ b2, float* __restrict__ kap) {
  int b = blockIdx.x * blockDim.x + threadIdx.x;
  if (b >= Bsz) return;
  float acc = b2[0];
  for (int j = 0; j < Hdim; j++) acc += h1[b * Hdim + j] * W2[j];
  float sp = (acc > 20.0f) ? acc : log1pf(expf(acc));
  kap[b] = sp + 1e-6f;
}

// ---------- 4) per-class prep: Q, R, scm_norms, bias ----------
__global__ void k_classprep(const float* __restrict__ Wmu, const float* __restrict__ Wk,
                            float* __restrict__ Q, float* __restrict__ R,
                            float* __restrict__ sn, float* __restrict__ bias) {
  __shared__ float sd[128];
  int c = blockIdx.x, d = threadIdx.x;
  if (c >= Cdim) {  // zero padding rows so padded GEMM tiles are inert
    Q[c * Fdim + d] = 0.0f; R[c * Fdim + d] = 0.0f;
    if (d == 0) { sn[c] = 0.0f; bias[c] = 0.0f; }
    return;
  }
  float mu = Wmu[c * Fdim + d];
  float kap = fmaxf(Wk[c * Fdim + d], 0.1f);
  float s1 = block_sum_128(mu * mu, sd);
  float mun = mu / fmaxf(sqrtf(s1), 1e-12f);
  float scm = mun * kap;
  float s2 = block_sum_128(scm * scm, sd);
  float z = sqrtf(s2);
  float zin = 1.0f / fmaxf(z, 1e-12f);
  Q[c * Fdim + d] = kap * scm * zin;   // kappa * nscm
  R[c * Fdim + d] = kap * kap;
  float s3 = block_sum_128(logf(kap), sd);
  if (d == 0) {
    const float nu = 63.0f;               // D/2 - 1
    const float L2PI = 1.8378770664093453f;
    float eta = sqrtf(z * z + nu * nu);
    float logiv = eta + nu * logf(z / (nu + eta)) - 0.5f * L2PI
                  - 0.25f * logf(z * z + nu * nu);
    float lognorm = nu * logf(z) - 64.0f * L2PI - logiv;  // D/2 = 64
    sn[c] = z;
    bias[c] = lognorm + (s3 - logf(z));   // + log_det
  }
}

// ---------- 5) vMF sampling: one wave per (s,b); lane owns dims lane+32j ----------
__global__ void k_sample(const float* __restrict__ mus, const float* __restrict__ kap,
                         float* __restrict__ S, float* __restrict__ P) {
  int b = blockIdx.x;                 // 0..63
  int s = threadIdx.x >> 5;           // 0..7 (8 waves per block)
  int lane = threadIdx.x & 31;
  float k = kap[b];
  const float m1 = 127.0f;
  float sq = sqrtf(4.0f * k * k + m1 * m1);
  float bb = (-2.0f * k + sq) / m1;
  float aa = (m1 + 2.0f * k + sq) * 0.25f;
  float dd = 4.0f * aa * bb / (1.0f + bb) - m1 * logf(m1);
  unsigned base = (unsigned)(s * Bsz + b);
  // rejection over K_ROUNDS (wave-uniform); Beta(63.5,63.5) ~ N(0.5, 0.5/sqrt(128))
  float sel = 0.0f, eps0 = 0.0f; bool done = false;
  for (int r = 0; r < 32; r++) {
    unsigned h = base * 2654435769u + (unsigned)r * 40503u;
    float u1 = u01(h + 1u), u2 = u01(h + 2u), uu = u01(h + 3u);
    float n = sqrtf(-2.0f * logf(u1)) * cosf(6.2831853f * u2);
    float eps = fminf(fmaxf(0.5f + 0.044194174f * n, 1e-6f), 1.0f - 1e-6f);
    if (r == 0) eps0 = eps;
    float t = 2.0f * aa * bb / (1.0f - (1.0f - bb) * eps);
    bool acc = (m1 * logf(t) - t + dd) >= logf(uu);
    if (acc && !done) { sel = eps; done = true; }
  }
  if (!done) sel = eps0;              // argmax over all-false -> index 0
  float w = (1.0f - (1.0f + bb) * sel) / (1.0f - (1.0f - bb) * sel);
  // tangential gaussian (dims 1..127), normalize
  float vloc[4]; float ss = 0.0f;
  #pragma unroll
  for (int j = 0; j < 4; j++) {
    int d = lane + 32 * j;
    float n = 0.0f;
    if (d >= 1) {
      unsigned h = (base * 131u + (unsigned)d) * 2246822519u;
      float u1 = u01(h + 11u), u2 = u01(h + 29u);
      n = sqrtf(-2.0f * logf(u1)) * cosf(6.2831853f * u2);
    }
    vloc[j] = n; ss += n * n;
  }
  ss = wave_sum(ss);
  float vinv = 1.0f / fmaxf(sqrtf(ss), 1e-12f);
  float wr = sqrtf(fmaxf(1.0f - w * w, 0.0f));
  // z and Householder direction uh = normalize(e1 - mu)
  float zloc[4], uhloc[4]; float un = 0.0f;
  #pragma unroll
  for (int j = 0; j < 4; j++) {
    int d = lane + 32 * j;
    zloc[j] = (d == 0) ? w : wr * vloc[j] * vinv;
    float mu = mus[b * Fdim + d];
    float uh = ((d == 0) ? 1.0f : 0.0f) - mu;
    uhloc[j] = uh; un += uh * uh;
  }
  un = wave_sum(un);
  float uinv = 1.0f / fmaxf(sqrtf(un), 1e-12f);
  float dot = 0.0f;
  #pragma unroll
  for (int j = 0; j < 4; j++) { uhloc[j] *= uinv; dot += zloc[j] * uhloc[j]; }
  dot = wave_sum(dot);
  int mrow = s * Bsz + b;
  #pragma unroll
  for (int j = 0; j < 4; j++) {
    int d = lane + 32 * j;
    float sv = zloc[j] - 2.0f * dot * uhloc[j];
    S[mrow * Fdim + d] = sv;
    P[mrow * Fdim + d] = sv * sv;
  }
}

// ---------- 6) fused dual-GEMM epilogue: out = bias + ||scm|| * (S·Qᵀ)/sqrt(P·Rᵀ) ----------
__global__ void k_vmf_gemm(const float* __restrict__ S, const float* __restrict__ P,
                           const float* __restrict__ Q, const float* __restrict__ R,
                           const float* __restrict__ sn, const float* __restrict__ bias,
                           float* __restrict__ Out) {
  const int K = Fdim, tilesN = Cpad / 16;            // 63
  int wave = (blockIdx.x * blockDim.x + threadIdx.x) >> 5;
  int lane = threadIdx.x & 31;
  int tm = wave / tilesN, tn = wave % tilesN;
  int m0 = tm * 16, n0 = tn * 16;
  if (m0 >= Mrows) return;
  int row = lane & 15, hi = lane >> 4;
  const float* sp = S + (m0 + row) * K + hi * 2;
  const float* pp = P + (m0 + row) * K + hi * 2;
  const float* qp = Q + (n0 + row) * K + hi * 2;
  const float* rp = R + (n0 + row) * K + hi * 2;
  v8f accA = {}, accN = {};
  for (int k0 = 0; k0 < K; k0 += 4) {
    v2f aS, aP, bQ, bR;
    aS.x = sp[k0]; aS.y = sp[k0 + 1];
    aP.x = pp[k0]; aP.y = pp[k0 + 1];
    bQ.x = qp[k0]; bQ.y = qp[k0 + 1];
    bR.x = rp[k0]; bR.y = rp[k0 + 1];
    accA = __builtin_amdgcn_wmma_f32_16x16x4_f32(false, aS, false, bQ, (short)0, accA,
                                                 false, false);
    accN = __builtin_amdgcn_wmma_f32_16x16x4_f32(false, aP, false, bR, (short)0, accN,
                                                 false, false);
  }
  int c = n0 + (lane & 15);
  if (c >= Cdim) return;              // divergence only after WMMA is done
  int mb = m0 + hi * 8;
  float snv = sn[c], bi = bias[c];
  #pragma unroll
  for (int v = 0; v < 8; v++) {
    float cosv = accA[v] / fmaxf(sqrtf(fmaxf(accN[v], 0.0f)), 1e-12f);
    Out[(mb + v) * Cdim + c] = bi + snv * cosv;
  }
}

extern "C" void kernel_launch(void* const* d_in, const int* in_sizes, int n_in,
                              void* d_out, int out_size, void* d_ws, size_t ws_size,
                              hipStream_t stream) {
  const float* features = (const float*)d_in[0];
  const float* W_mu     = (const float*)d_in[1];
  const float* W_kappa  = (const float*)d_in[2];
  const float* W0       = (const float*)d_in[3];
  const float* b0       = (const float*)d_in[4];
  const float* W1       = (const float*)d_in[5];
  const float* b1       = (const float*)d_in[6];
  const float* W2       = (const float*)d_in[7];
  const float* b2       = (const float*)d_in[8];
  float* out = (float*)d_out;

  float* ws   = (float*)d_ws;
  float* mus  = ws;                       // 64*128
  float* h0   = mus + Bsz * Fdim;         // 64*256
  float* h1   = h0 + Bsz * Hdim;          // 64*256
  float* kap  = h1 + Bsz * Hdim;          // 64
  float* Q    = kap + Bsz;                // 1008*128
  float* R    = Q + Cpad * Fdim;          // 1008*128
  float* sn   = R + Cpad * Fdim;          // 1008
  float* bias = sn + Cpad;                // 1008
  float* Smat = bias + Cpad;              // 512*128
  float* Pmat = Smat + Mrows * Fdim;      // 512*128

  // batch mus
  k_norm_rows<<<Bsz, 128, 0, stream>>>(features, mus);
  // MLP (uses raw features, per reference)
  k_mlp_wmma<<<8, 256, 0, stream>>>(features, W0, b0, h0, Bsz, Hdim, Fdim, Hdim / 16);
  k_mlp_wmma<<<8, 256, 0, stream>>>(h0, W1, b1, h1, Bsz, Hdim, Hdim, Hdim / 16);
  k_kappa<<<1, 64, 0, stream>>>(h1, W2, b2, kap);
  // class prep (padded to 1008)
  k_classprep<<<Cpad, 128, 0, stream>>>(W_mu, W_kappa, Q, R, sn, bias);
  // vMF sampling: 64 blocks x 8 waves (one wave per (s,b))
  k_sample<<<Bsz, 256, 0, stream>>>(mus, kap, Smat, Pmat);
  // dual GEMM + epilogue: 32 x 63 tiles, 8 waves/block
  {
    int waves = (Mrows / 16) * (Cpad / 16);   // 2016
    int blocks = waves / 8;                   // 252
    k_vmf_gemm<<<blocks, 256, 0, stream>>>(Smat, Pmat, Q, R, sn, bias, out);
  }
}